// PNA_regression_87282325390049
// MI455X (gfx1250) — compile-verified
//
#include <hip/hip_runtime.h>
#include <hip/hip_bf16.h>

typedef __attribute__((ext_vector_type(2))) float v2f;
typedef __attribute__((ext_vector_type(8))) float v8f;

#define PNA_ALPHA   0.2f
#define PNA_AVG_LIN 16.0f
#define PNA_AVG_LOG 2.8332133440562162f   // log(17.0)

// --------------------------------------------------------------------------
// Optional CDNA5 async global->LDS direct copy (ASYNCcnt path, ISA 15.18.3).
// Builtin signature (from hipcc diagnostic): takes pointers to
// 'int __attribute__((vector_size(16)))' in AS1 (global) and AS3 (LDS),
// plus constant offset and cpol immediates.
// --------------------------------------------------------------------------
#if defined(__gfx1250__) && __has_builtin(__builtin_amdgcn_global_load_async_to_lds_b128)
#define HAVE_ASYNC_LDS 1
#endif

#ifdef HAVE_ASYNC_LDS
typedef int async_v4i __attribute__((vector_size(16)));
typedef __attribute__((address_space(1))) async_v4i* GlobV4;
typedef __attribute__((address_space(3))) async_v4i* LdsV4;

__device__ __forceinline__ void async_copy16(const void* gptr, void* lptr) {
  // generic->AS pointers via integer cast: global generic == AS1 address,
  // LDS generic low 32 bits == LDS byte offset (aperture in high bits).
  __builtin_amdgcn_global_load_async_to_lds_b128(
      (GlobV4)(unsigned long long)gptr,
      (LdsV4)(unsigned int)(unsigned long long)lptr,
      0, 0);
}

__device__ __forceinline__ void async_wait0() {
#if __has_builtin(__builtin_amdgcn_s_wait_asynccnt)
  __builtin_amdgcn_s_wait_asynccnt(0);
#else
  asm volatile("s_wait_asynccnt 0" ::: "memory");
#endif
}
#endif

// ---------------------------------------------------------------------------
// WMMA helper: accumulate C[16x16] += A[16xK] * B[K x 64][:, cb:cb+16]
// A in LDS (row-major, stride lda, lda % 64 == 4 -> conflict-free),
// B in global memory (row-major, width 64, L2/L0 resident weights; compiler
// hoists all B loads into registers across the unrolled loop).
// Fragment layouts per CDNA5 ISA 7.12.2:
//   A(m,k): lane = m + 16*(k>>1), vgpr = k&1
//   B(k,n): lane = n + 16*(k>>1), vgpr = k&1
//   C(m,n): vgpr i -> m = i + 8*(lane>>4), n = lane&15
// ---------------------------------------------------------------------------
__device__ __forceinline__ v8f wmma_acc_k(v8f c, const float* __restrict__ As, int lda,
                                          const float* __restrict__ B, int cb,
                                          int K, int lane) {
  const int mn = lane & 15;          // row of A fragment == col of B fragment
  const int kh = (lane >> 4) << 1;   // 0 or 2: which pair of K this half-wave holds
  for (int k0 = 0; k0 < K; k0 += 4) {
    v2f a, b;
    a.x = As[mn * lda + k0 + kh];
    a.y = As[mn * lda + k0 + kh + 1];
    b.x = B[(k0 + kh)     * 64 + cb + mn];
    b.y = B[(k0 + kh + 1) * 64 + cb + mn];
    c = __builtin_amdgcn_wmma_f32_16x16x4_f32(false, a, false, b, (short)0, c, false, false);
  }
  return c;
}

// sign-aware float atomic max via int punning (mx buffer initialized to -inf)
__device__ __forceinline__ void atomicMaxF(float* addr, float v) {
  if (v >= 0.0f) atomicMax((int*)addr, __float_as_int(v));
  else           atomicMin((unsigned int*)addr, __float_as_uint(v));
}

// ---------------------------------------------------------------------------
// Layer 0: C = leaky_relu(A[M x 128] @ W[128 x 64] + bias)
// block = 128 threads (4 waves); block computes 16 rows x 64 cols
// ---------------------------------------------------------------------------
__global__ void gemm_k128_kernel(const float* __restrict__ A, const float* __restrict__ W,
                                 const float* __restrict__ bias, float* __restrict__ C,
                                 int M) {
  __shared__ float As[16 * 132];
  const int LDA = 132;
  const int tid = threadIdx.x;
  const int rowBase = blockIdx.x * 16;

#ifdef HAVE_ASYNC_LDS
  for (int idx = tid; idx < 512; idx += 128) {      // 16 rows x 32 chunks of 16B
    int r = idx >> 5, chunk = idx & 31;
    int gr = rowBase + r;
    float* lp = &As[r * LDA + (chunk << 2)];
    if (gr < M) {
      async_copy16(&A[(size_t)gr * 128 + (chunk << 2)], lp);
    } else {
      lp[0] = lp[1] = lp[2] = lp[3] = 0.0f;
    }
  }
  async_wait0();
#else
  for (int idx = tid; idx < 16 * 128; idx += 128) {
    int r = idx >> 7, c = idx & 127;
    int gr = rowBase + r;
    As[r * LDA + c] = (gr < M) ? A[(size_t)gr * 128 + c] : 0.0f;
  }
#endif
  __syncthreads();

  const int lane = tid & 31, wave = tid >> 5;
  const int cb = wave * 16;
  v8f acc = {};
  acc = wmma_acc_k(acc, As, LDA, W, cb, 128, lane);

  const int n = lane & 15, khi = lane >> 4;
  const float bv = bias[cb + n];
#pragma unroll
  for (int i = 0; i < 8; ++i) {
    int gr = rowBase + i + 8 * khi;
    if (gr < M) {
      float v = acc[i] + bv;
      v = (v > 0.0f) ? v : PNA_ALPHA * v;   // leaky relu
      C[(size_t)gr * 64 + cb + n] = v;
    }
  }
}

// ---------------------------------------------------------------------------
// Edge messages + fused aggregation.
// Per block: 16 edges. A tile [16 x 128] = [h[dst] | h[src]] gathered to LDS
// (async direct-to-LDS when available). msg = A @ preW + preB, then atomic
// scatter-add into sum and sign-aware atomic-max into mx (both L2-resident).
// ---------------------------------------------------------------------------
__global__ void edge_msg_kernel(const float* __restrict__ H,
                                const int* __restrict__ src, const int* __restrict__ dst,
                                const float* __restrict__ preW, const float* __restrict__ preB,
                                float* __restrict__ sumBuf, float* __restrict__ maxBuf,
                                int E) {
  __shared__ float As[16 * 132];
  __shared__ int sDst[16];
  __shared__ int sNode[32];   // [0:16) = dst nodes, [16:32) = src nodes
  const int LDA = 132;
  const int tid = threadIdx.x;
  const int ebase = blockIdx.x * 16;

  if (tid < 16) {
    int e = ebase + tid;
    int d = (e < E) ? dst[e] : -1;
    int s = (e < E) ? src[e] : -1;
    sDst[tid] = d;
    sNode[tid] = d;
    sNode[16 + tid] = s;
  }
  __syncthreads();

#ifdef HAVE_ASYNC_LDS
  for (int idx = tid; idx < 512; idx += 128) {      // 16 rows x 32 chunks of 16B
    int r = idx >> 5, chunk = idx & 31;             // chunk<16 -> dst half, else src half
    int node = sNode[((chunk >> 4) << 4) + r];
    float* lp = &As[r * LDA + (chunk << 2)];
    if (node >= 0) {
      async_copy16(&H[(size_t)node * 64 + ((chunk & 15) << 2)], lp);
    } else {
      lp[0] = lp[1] = lp[2] = lp[3] = 0.0f;
    }
  }
  async_wait0();
#else
  for (int idx = tid; idx < 16 * 128; idx += 128) {
    int r = idx >> 7, c = idx & 127;
    int node = sNode[((c >> 6) << 4) + r];          // c<64 -> dst row, else src row
    As[r * LDA + c] = (node >= 0) ? H[(size_t)node * 64 + (c & 63)] : 0.0f;
  }
#endif
  __syncthreads();

  const int lane = tid & 31, wave = tid >> 5;
  const int cb = wave * 16;
  v8f acc = {};
  acc = wmma_acc_k(acc, As, LDA, preW, cb, 128, lane);

  const int n = lane & 15, khi = lane >> 4;
  const float bv = preB[cb + n];
#pragma unroll
  for (int i = 0; i < 8; ++i) {
    int r = i + 8 * khi;
    int d = sDst[r];
    if (d >= 0) {
      float v = acc[i] + bv;
      size_t off = (size_t)d * 64 + cb + n;
      atomicAdd(&sumBuf[off], v);
      atomicMaxF(&maxBuf[off], v);
    }
  }
}

// ---------------------------------------------------------------------------
// Post: out = relu( (cat[x, att, lin] @ postW + postB) @ linW + linB )
// per block: 16 nodes; build [16 x 320] scaled tile in LDS, two chained GEMMs.
// ---------------------------------------------------------------------------
__global__ void post_kernel(const float* __restrict__ H,
                            const float* __restrict__ sumBuf, const float* __restrict__ maxBuf,
                            const float* __restrict__ cnt,
                            const float* __restrict__ postW, const float* __restrict__ postB,
                            const float* __restrict__ linW,  const float* __restrict__ linB,
                            float* __restrict__ Out, int N) {
  __shared__ float As[16 * 324];
  __shared__ float Tmp[16 * 68];
  __shared__ float sS1[16], sS2[16], sInv[16], sMsk[16];
  const int LDA = 324, LDT = 68;
  const int tid = threadIdx.x;
  const int rowBase = blockIdx.x * 16;

  if (tid < 16) {
    int gr = rowBase + tid;
    float c = (gr < N) ? cnt[gr] : 0.0f;
    float deg = fmaxf(c, 1.0f);
    sS1[tid] = PNA_AVG_LOG / logf(deg + 1.0f);
    sS2[tid] = deg / PNA_AVG_LIN;
    sInv[tid] = 1.0f / deg;
    sMsk[tid] = (c > 0.0f) ? 1.0f : 0.0f;
  }
  __syncthreads();

  {
    const int r = tid >> 3;            // 8 threads per row
    const int gr = rowBase + r;
    const bool valid = gr < N;
    const float s1 = sS1[r], s2 = sS2[r], inv = sInv[r], msk = sMsk[r];
    for (int c = (tid & 7); c < 320; c += 8) {
      float v = 0.0f;
      if (valid) {
        int j = c & 63;
        int seg = c >> 6;              // 0:x 1:mean*s1 2:mx*s1 3:mean*s2 4:mx*s2
        if (seg == 0) {
          v = H[(size_t)gr * 64 + j];
        } else {
          float mean = sumBuf[(size_t)gr * 64 + j] * inv;
          float mx = (msk > 0.0f) ? maxBuf[(size_t)gr * 64 + j] : 0.0f;
          float base = ((seg == 1) || (seg == 3)) ? mean : mx;
          v = base * ((seg <= 2) ? s1 : s2);
        }
      }
      As[r * LDA + c] = v;
    }
  }
  __syncthreads();

  const int lane = tid & 31, wave = tid >> 5;
  const int cb = wave * 16;
  const int n = lane & 15, khi = lane >> 4;

  v8f acc = {};
  acc = wmma_acc_k(acc, As, LDA, postW, cb, 320, lane);
  {
    const float bv = postB[cb + n];
#pragma unroll
    for (int i = 0; i < 8; ++i)
      Tmp[(i + 8 * khi) * LDT + cb + n] = acc[i] + bv;
  }
  __syncthreads();

  v8f acc2 = {};
  acc2 = wmma_acc_k(acc2, Tmp, LDT, linW, cb, 64, lane);
  const float bv2 = linB[cb + n];
#pragma unroll
  for (int i = 0; i < 8; ++i) {
    int gr = rowBase + i + 8 * khi;
    if (gr < N) Out[(size_t)gr * 64 + cb + n] = fmaxf(acc2[i] + bv2, 0.0f);  // relu
  }
}

// ---------------------------------------------------------------------------
// Small utility kernels
// ---------------------------------------------------------------------------
__global__ void zero_kernel(float* __restrict__ p, int n) {
  int i = blockIdx.x * blockDim.x + threadIdx.x;
  if (i < n) p[i] = 0.0f;
}

__global__ void init_aggr_kernel(float* __restrict__ s, float* __restrict__ m, int n) {
  int i = blockIdx.x * blockDim.x + threadIdx.x;
  if (i < n) { s[i] = 0.0f; m[i] = __int_as_float(0xFF800000); }  // -inf
}

__global__ void count_kernel(const int* __restrict__ dst, float* __restrict__ cnt, int E) {
  int e = blockIdx.x * blockDim.x + threadIdx.x;
  if (e < E) atomicAdd(&cnt[dst[e]], 1.0f);
}

// final: out[n] = dot(H[n,:], W2) + b2, one wave32 per node
__global__ void final_kernel(const float* __restrict__ H, const float* __restrict__ W2,
                             const float* __restrict__ b2, float* __restrict__ out, int N) {
  int lane = threadIdx.x & 31;
  int node = (blockIdx.x * blockDim.x + threadIdx.x) >> 5;
  if (node >= N) return;
  float v = H[(size_t)node * 64 + lane] * W2[lane]
          + H[(size_t)node * 64 + 32 + lane] * W2[32 + lane];
#pragma unroll
  for (int off = 16; off; off >>= 1) v += __shfl_xor(v, off, 32);
  if (lane == 0) out[node] = v + b2[0];
}

// ---------------------------------------------------------------------------
extern "C" void kernel_launch(void* const* d_in, const int* in_sizes, int n_in,
                              void* d_out, int out_size, void* d_ws, size_t ws_size,
                              hipStream_t stream) {
  const float* x         = (const float*)d_in[0];
  const int*   edge_idx  = (const int*)  d_in[2];
  const float* W0        = (const float*)d_in[3];
  const float* b0        = (const float*)d_in[4];
  const float* c1_pre_W  = (const float*)d_in[5];
  const float* c1_pre_b  = (const float*)d_in[6];
  const float* c1_post_W = (const float*)d_in[7];
  const float* c1_post_b = (const float*)d_in[8];
  const float* c1_lin_W  = (const float*)d_in[9];
  const float* c1_lin_b  = (const float*)d_in[10];
  const float* c2_pre_W  = (const float*)d_in[11];
  const float* c2_pre_b  = (const float*)d_in[12];
  const float* c2_post_W = (const float*)d_in[13];
  const float* c2_post_b = (const float*)d_in[14];
  const float* c2_lin_W  = (const float*)d_in[15];
  const float* c2_lin_b  = (const float*)d_in[16];
  const float* W2        = (const float*)d_in[17];
  const float* b2        = (const float*)d_in[18];

  const int N = in_sizes[0] / 128;
  const int E = in_sizes[2] / 2;
  const int* src = edge_idx;
  const int* dst = edge_idx + E;

  float* h0     = (float*)d_ws;
  float* h1     = h0     + (size_t)N * 64;
  float* sumBuf = h1     + (size_t)N * 64;
  float* maxBuf = sumBuf + (size_t)N * 64;
  float* cntBuf = maxBuf + (size_t)N * 64;

  const int nTiles = (N + 15) / 16;
  const int eTiles = (E + 15) / 16;
  const int nvec   = N * 64;

  // layer 0: h0 = leaky_relu(x @ W0 + b0)
  gemm_k128_kernel<<<nTiles, 128, 0, stream>>>(x, W0, b0, h0, N);

  // per-node in-degree counts (shared by both convs)
  zero_kernel<<<(N + 255) / 256, 256, 0, stream>>>(cntBuf, N);
  count_kernel<<<(E + 255) / 256, 256, 0, stream>>>(dst, cntBuf, E);

  // conv 1: h0 -> h1
  init_aggr_kernel<<<(nvec + 255) / 256, 256, 0, stream>>>(sumBuf, maxBuf, nvec);
  edge_msg_kernel<<<eTiles, 128, 0, stream>>>(h0, src, dst, c1_pre_W, c1_pre_b,
                                              sumBuf, maxBuf, E);
  post_kernel<<<nTiles, 128, 0, stream>>>(h0, sumBuf, maxBuf, cntBuf,
                                          c1_post_W, c1_post_b, c1_lin_W, c1_lin_b, h1, N);

  // conv 2: h1 -> h0
  init_aggr_kernel<<<(nvec + 255) / 256, 256, 0, stream>>>(sumBuf, maxBuf, nvec);
  edge_msg_kernel<<<eTiles, 128, 0, stream>>>(h1, src, dst, c2_pre_W, c2_pre_b,
                                              sumBuf, maxBuf, E);
  post_kernel<<<nTiles, 128, 0, stream>>>(h1, sumBuf, maxBuf, cntBuf,
                                          c2_post_W, c2_post_b, c2_lin_W, c2_lin_b, h0, N);

  // final: out = h0 @ W2 + b2
  final_kernel<<<(N + 7) / 8, 256, 0, stream>>>(h0, W2, b2, (float*)d_out, N);
}